// ResidualVQ_38242388804089
// MI455X (gfx1250) — compile-verified
//
#include <hip/hip_runtime.h>
#include <hip/hip_bf16.h>

typedef __attribute__((ext_vector_type(2))) float v2f;
typedef __attribute__((ext_vector_type(8))) float v8f;
typedef unsigned int u32x4 __attribute__((ext_vector_type(4)));
typedef int i32x4 __attribute__((ext_vector_type(4)));
typedef int i32x8 __attribute__((ext_vector_type(8)));

// D = A(16x4) * B(4x16) + C(16x16), all f32.  CDNA5 V_WMMA_F32_16X16X4_F32.
__device__ __forceinline__ v8f wmma_f32(v2f a, v2f b, v8f c) {
  return __builtin_amdgcn_wmma_f32_16x16x4_f32(false, a, false, b, (short)0, c,
                                               false, false);
}

// ---------------------------------------------------------------------------
// TDM: DMA a 2D tile (64 rows x 768 f32, row stride 24576 elements) from
// global memory into LDS.  D# built per CDNA5 ISA 8.3/8.4.
// ---------------------------------------------------------------------------
__device__ __forceinline__ void tdm_load_xw_slice(const float* xw, int t,
                                                  unsigned lds_off) {
  unsigned long long ga =
      (unsigned long long)(const void*)(xw + (long long)t * 768);
  u32x4 g0;
  g0[0] = 1u;                                          // count=1, user mode
  g0[1] = lds_off;                                     // lds_addr (bytes)
  g0[2] = (unsigned)ga;                                // global_addr[31:0]
  g0[3] = (unsigned)((ga >> 32) & 0x1FFFFFFull) | (2u << 30);  // addr|type=2
  i32x8 g1;
  g1[0] = (int)(2u << 16);    // workgroup_mask=0, data_size=2 (4 bytes)
  g1[1] = (int)(768u << 16);  // tensor_dim0[15:0] << 16
  g1[2] = (int)(64u << 16);   // tensor_dim0[31:16]=0 | tensor_dim1[15:0]<<16
  g1[3] = (int)(768u << 16);  // tensor_dim1[31:16]=0 | tile_dim0<<16
  g1[4] = 64;                 // tile_dim1 | tile_dim2<<16
  g1[5] = 24576;              // tensor_dim0_stride[31:0] (32*768 elements)
  g1[6] = 0;                  // stride0 hi | stride1 lo
  g1[7] = 0;
  i32x4 z = {0, 0, 0, 0};
#if __clang_major__ >= 23
  i32x8 z8 = {0, 0, 0, 0, 0, 0, 0, 0};
  __builtin_amdgcn_tensor_load_to_lds(g0, g1, z, z, z8, 0);
#else
  __builtin_amdgcn_tensor_load_to_lds(g0, g1, z, z, 0);
#endif
}

// ---------------------------------------------------------------------------
// conv0: Cin=1, 3x3, stride 2, pad 1.  (64,1,2000,80) -> (64,32,1000,40)
// ---------------------------------------------------------------------------
__global__ void conv0_kernel(const float* __restrict__ x,
                             const float* __restrict__ w,
                             float* __restrict__ y) {
  const int OH = 1000, OW = 40, H = 2000, W = 80, C = 32;
  const long long total = 64LL * C * OH * OW;
  for (long long i = blockIdx.x * (long long)blockDim.x + threadIdx.x;
       i < total; i += (long long)gridDim.x * blockDim.x) {
    int ow = (int)(i % OW);
    long long t = i / OW;
    int oh = (int)(t % OH); t /= OH;
    int c = (int)(t % C);
    int b = (int)(t / C);
    int ih0 = oh * 2 - 1, iw0 = ow * 2 - 1;
    float acc = 0.f;
#pragma unroll
    for (int kh = 0; kh < 3; ++kh) {
      int ih = ih0 + kh;
      if (ih < 0 || ih >= H) continue;
#pragma unroll
      for (int kw = 0; kw < 3; ++kw) {
        int iw = iw0 + kw;
        if (iw < 0 || iw >= W) continue;
        acc += x[((long long)b * H + ih) * W + iw] * w[c * 9 + kh * 3 + kw];
      }
    }
    y[i] = acc;
  }
}

// ---------------------------------------------------------------------------
// Implicit-GEMM WMMA conv: M=Cout, N=B*OH*OW, K=Cin*9, stride 2, pad 1.
// ---------------------------------------------------------------------------
__global__ void conv_wmma_kernel(const float* __restrict__ x,
                                 const float* __restrict__ w,
                                 float* __restrict__ y,
                                 int N, int Cin, int H, int W,
                                 int Cout, int OH, int OW) {
  const int K = Cin * 9;
  const int npos = N * OH * OW;
  const int mtiles = Cout >> 4;
  const int ntiles = npos >> 4;
  int lane = threadIdx.x & 31;
  int wave = blockIdx.x * (blockDim.x >> 5) + (threadIdx.x >> 5);
  if (wave >= mtiles * ntiles) return;  // wave-uniform exit, EXEC stays full
  int mt = wave % mtiles, nt = wave / mtiles;
  int hl = lane >> 4, ln = lane & 15;

  int n = nt * 16 + ln;
  int ow = n % OW;
  int t = n / OW;
  int oh = t % OH;
  int b = t / OH;
  int ih0 = oh * 2 - 1, iw0 = ow * 2 - 1;

  const float* wrow = w + (long long)(mt * 16 + ln) * K;
  v8f acc = {};
  for (int k0 = 0; k0 < K; k0 += 4) {
    if ((k0 & 28) == 0 && k0 + 32 < K)  // global_prefetch_b8 ahead of weights
      __builtin_prefetch(wrow + k0 + 32, 0, 0);
    int kk = k0 + (hl << 1);
    v2f a;
    a.x = wrow[kk];
    a.y = wrow[kk + 1];
    v2f bb;
#pragma unroll
    for (int j = 0; j < 2; ++j) {
      int k = kk + j;
      int ci = k / 9;
      int r = k - ci * 9;
      int kh = r / 3;
      int kw = r - kh * 3;
      int ih = ih0 + kh, iw = iw0 + kw;
      float v = 0.f;
      if (ih >= 0 && ih < H && iw >= 0 && iw < W)
        v = x[(((long long)b * Cin + ci) * H + ih) * W + iw];
      if (j == 0) bb.x = v; else bb.y = v;
    }
    acc = wmma_f32(a, bb, acc);
  }
#pragma unroll
  for (int v = 0; v < 8; ++v) {
    int m = mt * 16 + v + (hl << 3);
    y[(((long long)b * Cout + m) * OH + oh) * OW + ow] = acc[v];
  }
}

// ---------------------------------------------------------------------------
// BatchNorm: per-(b,c) plane partial sums -> channel stats -> apply+ReLU
// ---------------------------------------------------------------------------
__global__ void bn_stats_kernel(const float* __restrict__ y,
                                float* __restrict__ stats, int C, int plane) {
  int bc = blockIdx.x;
  int c = bc % C;
  const float* p = y + (long long)bc * plane;
  float s = 0.f, q = 0.f;
  for (int i = threadIdx.x; i < plane; i += blockDim.x) {
    float v = p[i];
    s += v;
    q += v * v;
  }
  __shared__ float ss[256], qq[256];
  ss[threadIdx.x] = s;
  qq[threadIdx.x] = q;
  __syncthreads();
  for (int off = blockDim.x >> 1; off > 0; off >>= 1) {
    if (threadIdx.x < off) {
      ss[threadIdx.x] += ss[threadIdx.x + off];
      qq[threadIdx.x] += qq[threadIdx.x + off];
    }
    __syncthreads();
  }
  if (threadIdx.x == 0) {
    atomicAdd(&stats[c], ss[0]);
    atomicAdd(&stats[C + c], qq[0]);
  }
}

__global__ void bn_finalize_kernel(const float* __restrict__ stats,
                                   const float* __restrict__ g,
                                   const float* __restrict__ beta,
                                   float* __restrict__ scsh, int C,
                                   float invcnt) {
  int c = threadIdx.x;
  if (c < C) {
    float mu = stats[c] * invcnt;
    float var = stats[C + c] * invcnt - mu * mu;
    float sc = g[c] * rsqrtf(var + 1e-5f);
    scsh[c] = sc;
    scsh[C + c] = beta[c] - mu * sc;
  }
}

__global__ void bn_apply_relu_kernel(float* __restrict__ y,
                                     const float* __restrict__ scsh, int C,
                                     int plane, long long total) {
  for (long long i = blockIdx.x * (long long)blockDim.x + threadIdx.x;
       i < total; i += (long long)gridDim.x * blockDim.x) {
    int c = (int)((i / plane) % C);
    float v = y[i] * scsh[c] + scsh[C + c];
    y[i] = v > 0.f ? v : 0.f;
  }
}

// ---------------------------------------------------------------------------
// (64,128,32,2) NCHW -> (64,32,256) [b][t][c*2+f]
// ---------------------------------------------------------------------------
__global__ void to_hs_kernel(const float* __restrict__ y5,
                             float* __restrict__ hs) {
  int i = blockIdx.x * blockDim.x + threadIdx.x;
  if (i >= 64 * 32 * 256) return;
  int d = i & 255;
  int t = (i >> 8) & 31;
  int b = i >> 13;
  int c = d >> 1, f = d & 1;
  hs[i] = y5[(((b * 128 + c) * 32) + t) * 2 + f];
}

// ---------------------------------------------------------------------------
// xw = hs(2048x256) @ w_ih^T(256x768) + bih : f32 WMMA GEMM
// ---------------------------------------------------------------------------
__global__ void gemm_xw_kernel(const float* __restrict__ A,
                               const float* __restrict__ Wt,
                               const float* __restrict__ bias,
                               float* __restrict__ Cm, int M, int N, int K) {
  int lane = threadIdx.x & 31;
  int wave = blockIdx.x * (blockDim.x >> 5) + (threadIdx.x >> 5);
  int mtiles = M >> 4, ntiles = N >> 4;
  if (wave >= mtiles * ntiles) return;
  int mt = wave % mtiles, nt = wave / mtiles;
  int hl = lane >> 4, ln = lane & 15;
  const float* arow = A + (long long)(mt * 16 + ln) * K;
  const float* brow = Wt + (long long)(nt * 16 + ln) * K;  // B(k,n)=Wt[n*K+k]
  v8f acc = {};
  for (int k0 = 0; k0 < K; k0 += 4) {
    int kk = k0 + (hl << 1);
    v2f a, b;
    a.x = arow[kk];
    a.y = arow[kk + 1];
    b.x = brow[kk];
    b.y = brow[kk + 1];
    acc = wmma_f32(a, b, acc);
  }
  int n = nt * 16 + ln;
  float bn = bias[n];
#pragma unroll
  for (int v = 0; v < 8; ++v) {
    int m = mt * 16 + v + (hl << 3);
    Cm[(long long)m * N + n] = acc[v] + bn;
  }
}

// ---------------------------------------------------------------------------
// Persistent GRU: one workgroup (32 waves), h in LDS (64x256 f32), and the
// per-step xw slice (64x768 f32) streamed into LDS by the Tensor Data Mover,
// double-phased against compute (issue slice t+1 once slice t's readers pass
// the barrier).  Sync via s_wait_tensorcnt + workgroup barrier.
// ---------------------------------------------------------------------------
__global__ void __launch_bounds__(1024) gru_kernel(
    const float* __restrict__ xw, const float* __restrict__ whh,
    const float* __restrict__ bhh, float* __restrict__ hT) {
  extern __shared__ float hsm[];          // [0, 64*256)  : h
  float* xslice = hsm + 64 * 256;         // [.., +64*768): xw slice (TDM dest)
  const int Hd = 256, B = 64, T = 32;
  for (int i = threadIdx.x; i < B * Hd; i += blockDim.x) hsm[i] = 0.f;
  __syncthreads();
  int lane = threadIdx.x & 31;
  int wave = threadIdx.x >> 5;  // 0..31
  int hl = lane >> 4, ln = lane & 15;
  unsigned lds_off = (unsigned)(unsigned long long)(void*)xslice;

  if (threadIdx.x < 32) tdm_load_xw_slice(xw, 0, lds_off);  // prologue DMA

  for (int t = 0; t < T; ++t) {
    if (threadIdx.x < 32) __builtin_amdgcn_s_wait_tensorcnt(0);
    __syncthreads();  // slice t resident in LDS for all waves

    float hnew[2][8];
#pragma unroll
    for (int ti = 0; ti < 2; ++ti) {
      int task = wave * 2 + ti;  // 64 tasks = 4 m-tiles x 16 j-tiles
      int mt = task >> 4;
      int jt = task & 15;
      int j = jt * 16 + ln;
      const float* hrow = hsm + (mt * 16 + ln) * Hd;
      const float* wR = whh + (long long)j * Hd;
      const float* wZ = whh + (long long)(256 + j) * Hd;
      const float* wN = whh + (long long)(512 + j) * Hd;
      v8f aR = {}, aZ = {}, aN = {};
      for (int k0 = 0; k0 < Hd; k0 += 4) {
        int kk = k0 + (hl << 1);
        v2f a, b;
        a.x = hrow[kk];
        a.y = hrow[kk + 1];
        b.x = wR[kk]; b.y = wR[kk + 1]; aR = wmma_f32(a, b, aR);
        b.x = wZ[kk]; b.y = wZ[kk + 1]; aZ = wmma_f32(a, b, aZ);
        b.x = wN[kk]; b.y = wN[kk + 1]; aN = wmma_f32(a, b, aN);
      }
      float bR = bhh[j], bZ = bhh[256 + j], bN = bhh[512 + j];
#pragma unroll
      for (int v = 0; v < 8; ++v) {
        int bb = mt * 16 + v + (hl << 3);
        const float* xrow = xslice + bb * 768;  // staged by TDM
        float r = 1.f / (1.f + expf(-(xrow[j] + aR[v] + bR)));
        float z = 1.f / (1.f + expf(-(xrow[256 + j] + aZ[v] + bZ)));
        float n = tanhf(xrow[512 + j] + r * (aN[v] + bN));
        float hold = hsm[bb * Hd + j];
        hnew[ti][v] = (1.f - z) * n + z * hold;
      }
    }
    __syncthreads();  // everyone done reading h and xslice

    if (threadIdx.x < 32 && t + 1 < T)
      tdm_load_xw_slice(xw, t + 1, lds_off);  // overlap DMA with h update

#pragma unroll
    for (int ti = 0; ti < 2; ++ti) {
      int task = wave * 2 + ti;
      int mt = task >> 4, jt = task & 15;
      int j = jt * 16 + ln;
#pragma unroll
      for (int v = 0; v < 8; ++v)
        hsm[(mt * 16 + v + (hl << 3)) * Hd + j] = hnew[ti][v];
    }
    __syncthreads();
  }
  for (int i = threadIdx.x; i < B * Hd; i += blockDim.x) hT[i] = hsm[i];
}

// ---------------------------------------------------------------------------
// Persistent 3-stage residual VQ: z in LDS; scores via WMMA (||z||^2 dropped
// for argmin); packed (score,idx) ds_min_u64 argmin; zq1+zq2+zq3 = z0 - zfin.
// ---------------------------------------------------------------------------
__global__ void __launch_bounds__(1024) vq_kernel(
    const float* __restrict__ z0, const float* __restrict__ cb1,
    const float* __restrict__ cb2, const float* __restrict__ cb3,
    float* __restrict__ out) {
  extern __shared__ float sm[];
  float* z = sm;                                                // 64*256
  float* en = sm + 64 * 256;                                    // 1024
  unsigned long long* keys = (unsigned long long*)(en + 1024);  // 64
  float* lossp = (float*)(keys + 64);                           // 1
  const int B = 64, D = 256, NE = 1024;
  const long long O_LOSS = 49152, O_IDX = 49153, O_CB = 49217;

  for (int i = threadIdx.x; i < B * D; i += blockDim.x) z[i] = z0[i];
  if (threadIdx.x == 0) *lossp = 0.f;
  __syncthreads();

  int lane = threadIdx.x & 31;
  int wave = threadIdx.x >> 5;
  int hl = lane >> 4, ln = lane & 15;
  const float* cbs[3] = {cb1, cb2, cb3};

  for (int s = 0; s < 3; ++s) {
    const float* E = cbs[s];
    for (int e = threadIdx.x; e < NE; e += blockDim.x) {
      const float* er = E + (long long)e * D;
      float q = 0.f;
      for (int k = 0; k < D; ++k) q += er[k] * er[k];
      en[e] = q;
    }
    for (int i = threadIdx.x; i < B; i += blockDim.x) keys[i] = ~0ULL;
    __syncthreads();

#pragma unroll 1
    for (int ti = 0; ti < 8; ++ti) {  // 256 tiles / 32 waves
      int task = wave * 8 + ti;
      int mt = task & 3, et = task >> 2;
      const float* zrow = z + (mt * 16 + ln) * D;
      const float* erow = E + (long long)(et * 16 + ln) * D;
      v8f acc = {};
      for (int k0 = 0; k0 < D; k0 += 4) {
        int kk = k0 + (hl << 1);
        v2f a, b;
        a.x = zrow[kk];
        a.y = zrow[kk + 1];
        b.x = erow[kk];
        b.y = erow[kk + 1];
        acc = wmma_f32(a, b, acc);
      }
      int e = et * 16 + ln;
      float enorm = en[e];
#pragma unroll
      for (int v = 0; v < 8; ++v) {
        int bb = mt * 16 + v + (hl << 3);
        float score = enorm - 2.f * acc[v];
        unsigned u = __float_as_uint(score);
        u = (u & 0x80000000u) ? ~u : (u | 0x80000000u);  // order-preserving
        unsigned long long key = ((unsigned long long)u << 32) | (unsigned)e;
        atomicMin(&keys[bb], key);
      }
    }
    __syncthreads();

    float lsum = 0.f;
    for (int i = threadIdx.x; i < B * D; i += blockDim.x) {
      int bb = i >> 8, d = i & 255;
      int idx = (int)(keys[bb] & 0xFFFFFFFFULL);
      float q = E[(long long)idx * D + d];
      float zn = z[i] - q;
      z[i] = zn;
      lsum += zn * zn;
      out[(long long)bb * 768 + s * 256 + d] = q;  // z_q_out column block
      out[O_CB + (long long)s * 16384 + i] = q;    // codebooks[s]
    }
    atomicAdd(lossp, lsum * 1.25f / (float)(B * D));
    if (s == 0)
      for (int bb = threadIdx.x; bb < B; bb += blockDim.x)
        out[O_IDX + bb] = (float)(int)(keys[bb] & 0xFFFFFFFFULL);
    __syncthreads();
  }
  // zq1+zq2+zq3 == z0 - z_final  (avoids global RAW across waves)
  for (int i = threadIdx.x; i < B * D; i += blockDim.x)
    out[O_CB + 3LL * 16384 + i] = z0[i] - z[i];
  if (threadIdx.x == 0) out[O_LOSS] = *lossp;
}

// ---------------------------------------------------------------------------
// Host
// ---------------------------------------------------------------------------
extern "C" void kernel_launch(void* const* d_in, const int* in_sizes, int n_in,
                              void* d_out, int out_size, void* d_ws,
                              size_t ws_size, hipStream_t stream) {
  (void)in_sizes; (void)n_in; (void)out_size; (void)ws_size;
  const float* speech = (const float*)d_in[0];
  const float* cw[6]; const float* g[6]; const float* bt[6];
  for (int i = 0; i < 6; ++i) {
    cw[i] = (const float*)d_in[1 + 3 * i];
    g[i] = (const float*)d_in[2 + 3 * i];
    bt[i] = (const float*)d_in[3 + 3 * i];
  }
  const float* w_ih = (const float*)d_in[19];
  const float* w_hh = (const float*)d_in[20];
  const float* bih = (const float*)d_in[21];
  const float* bhh = (const float*)d_in[22];
  const float* cb1 = (const float*)d_in[23];
  const float* cb2 = (const float*)d_in[24];
  const float* cb3 = (const float*)d_in[25];
  float* out = (float*)d_out;

  char* ws = (char*)d_ws;
  float* bufA = (float*)ws;                          // 327,680,000 B
  float* bufB = (float*)(ws + 327680000);            // 81,920,000 B
  float* hs = (float*)(ws + 327680000 + 81920000);   // 2,097,152 B
  float* xw = (float*)(ws + 327680000 + 81920000 + 2097152);   // 6,291,456 B
  float* hT = (float*)(ws + 327680000 + 81920000 + 2097152 + 6291456);
  float* stats = (float*)((char*)hT + 65536);        // 1 KB
  float* scsh = (float*)((char*)stats + 1024);       // 1 KB

  // ---- conv0 (direct) ----
  conv0_kernel<<<32768, 256, 0, stream>>>(speech, cw[0], bufA);

  struct L { int N, Cin, H, W, Cout, OH, OW; };
  const L layers[6] = {{64, 1, 2000, 80, 32, 1000, 40},
                       {64, 32, 1000, 40, 32, 500, 20},
                       {64, 32, 500, 20, 64, 250, 10},
                       {64, 64, 250, 10, 64, 125, 5},
                       {64, 64, 125, 5, 128, 63, 3},
                       {64, 128, 63, 3, 128, 32, 2}};
  float* bufs[2] = {bufA, bufB};
  for (int l = 0; l < 6; ++l) {
    const L& p = layers[l];
    float* y = bufs[l & 1];
    if (l > 0) {
      const float* xin = bufs[(l - 1) & 1];
      int tiles = (p.Cout >> 4) * ((p.N * p.OH * p.OW) >> 4);
      int blocks = (tiles + 7) / 8;
      conv_wmma_kernel<<<blocks, 256, 0, stream>>>(
          xin, cw[l], y, p.N, p.Cin, p.H, p.W, p.Cout, p.OH, p.OW);
    }
    int plane = p.OH * p.OW;
    long long total = (long long)p.N * p.Cout * plane;
    hipMemsetAsync(stats, 0, 2 * p.Cout * sizeof(float), stream);
    bn_stats_kernel<<<p.N * p.Cout, 256, 0, stream>>>(y, stats, p.Cout, plane);
    bn_finalize_kernel<<<1, 128, 0, stream>>>(stats, g[l], bt[l], scsh, p.Cout,
                                              1.f / (float)(p.N * plane));
    bn_apply_relu_kernel<<<4096, 256, 0, stream>>>(y, scsh, p.Cout, plane,
                                                   total);
  }

  // ---- reshape to GRU input ----
  to_hs_kernel<<<2048, 256, 0, stream>>>(bufs[5 & 1], hs);

  // ---- xw = hs @ w_ih^T + bih  (2048x768x256) ----
  gemm_xw_kernel<<<768, 256, 0, stream>>>(hs, w_ih, bih, xw, 2048, 768, 256);

  // ---- persistent GRU: h (64 KB) + TDM-staged xw slice (192 KB) in LDS ----
  size_t gru_lds = 64 * 256 * sizeof(float) + 64 * 768 * sizeof(float);
  gru_kernel<<<1, 1024, gru_lds, stream>>>(xw, w_hh, bhh, hT);

  // ---- persistent residual VQ ----
  size_t vq_lds = 64 * 256 * sizeof(float) + 1024 * sizeof(float) +
                  64 * sizeof(unsigned long long) + 64;
  vq_kernel<<<1, 1024, vq_lds, stream>>>(hT, cb1, cb2, cb3, out);
}